// NodePooling_1726576857256
// MI455X (gfx1250) — compile-verified
//
#include <hip/hip_runtime.h>

typedef __attribute__((ext_vector_type(2))) float v2f;
typedef __attribute__((ext_vector_type(8))) float v8f;

#define G_MAX          1024
#define DP             1024   // D*P floats per node row (contiguous memory order p*256+d)
#define TILES_PER_WAVE 8      // 8 waves * 8 tiles * 16 cols = 1024 cols

// One-block exclusive prefix sum of counts -> offs (G <= 1024)
__global__ void __launch_bounds__(G_MAX)
scan_counts_kernel(const int* __restrict__ counts, int* __restrict__ offs, int g) {
    __shared__ int s[G_MAX];
    const int t = threadIdx.x;
    const int c = (t < g) ? counts[t] : 0;
    s[t] = c;
    __syncthreads();
    #pragma unroll
    for (int d = 1; d < G_MAX; d <<= 1) {
        int v = s[t];
        if (t >= d) v += s[t - d];
        __syncthreads();
        s[t] = v;
        __syncthreads();
    }
    if (t < g) offs[t] = s[t] - c;   // exclusive scan
}

// One block per graph segment. Segment-mean via ones-vector WMMA rank reduction:
// D = ones(16x4) * B(4x16) + C  => every row of D accumulates 4 node rows.
__global__ void __launch_bounds__(256)
seg_mean_wmma_kernel(const float* __restrict__ feat,
                     const int*   __restrict__ counts,
                     const int*   __restrict__ offs,
                     float*       __restrict__ out) {
    const int g     = blockIdx.x;
    const int cnt   = counts[g];
    const int start = offs[g];

    const int tid  = threadIdx.x;
    const int wave = tid >> 5;
    const int lane = tid & 31;
    const int half = lane >> 4;   // lanes 0-15: B rows K=0,1 ; lanes 16-31: K=2,3
    const int nl   = lane & 15;   // column within 16-wide tile

    const float* __restrict__ base = feat + (size_t)start * DP;
    const int wbase   = wave * (TILES_PER_WAVE * 16);  // this wave's first column
    const int colbase = wbase + nl;                    // per-lane column
    const int rowsel  = (2 * half) * DP;               // half-wave row-pair select

    v8f acc[TILES_PER_WAVE] = {};                      // zero C accumulators
    const v2f ones = {1.0f, 1.0f};                     // all-ones 16x4 A matrix

    auto loadStep = [&](int kk, v2f (&bb)[TILES_PER_WAVE]) {
        const int off = kk * DP + rowsel + colbase;
        #pragma unroll
        for (int t = 0; t < TILES_PER_WAVE; ++t) {
            bb[t].x = base[off + t * 16];
            bb[t].y = base[off + DP + t * 16];
        }
    };
    auto pfStep = [&](int kk) {                        // rows kk..kk+3 (both halves)
        const int pfo = kk * DP + rowsel + colbase;
        #pragma unroll
        for (int t = 0; t < TILES_PER_WAVE; ++t) {
            __builtin_prefetch(&base[pfo + t * 16], 0, 1);       // global_prefetch_b8
            __builtin_prefetch(&base[pfo + DP + t * 16], 0, 1);
        }
    };
    auto mmaStep = [&](v2f (&bb)[TILES_PER_WAVE]) {
        #pragma unroll
        for (int t = 0; t < TILES_PER_WAVE; ++t)
            acc[t] = __builtin_amdgcn_wmma_f32_16x16x4_f32(
                false, ones, false, bb[t], (short)0, acc[t], false, false);
    };

    v2f A[TILES_PER_WAVE], B[TILES_PER_WAVE];
    int k0 = 0;

    // Invariant: whenever k0 + 4 <= cnt outside the loop, A holds step k0.
    if (cnt >= 4) loadStep(0, A);

    // ---- main loop: 2x-unrolled software pipeline, no buffer rotation ----
    for (; k0 + 12 <= cnt; k0 += 8) {
        loadStep(k0 + 4, B);                           // loads for step i+1
        if (k0 + 20 <= cnt) pfStep(k0 + 16);           // uniform scalar guard
        mmaStep(A);                                    // consume step i
        loadStep(k0 + 8, A);                           // loads for step i+2
        if (k0 + 24 <= cnt) pfStep(k0 + 20);
        mmaStep(B);                                    // consume step i+1
    }

    // ---- drain: 0, 1 or 2 full steps remain ----
    if (k0 + 8 <= cnt) {
        loadStep(k0 + 4, B);
        mmaStep(A);
        mmaStep(B);
        k0 += 8;
    } else if (k0 + 4 <= cnt) {
        mmaStep(A);
        k0 += 4;
    }

    // ---- predicated tail: at most one partial k-step (1..3 rows) ----
    if (k0 < cnt) {
        const int  n0  = k0 + 2 * half;
        const bool p0  = (n0     < cnt);
        const bool p1  = (n0 + 1 < cnt);
        const int  off = k0 * DP + rowsel + colbase;
        #pragma unroll
        for (int t = 0; t < TILES_PER_WAVE; ++t) {
            v2f bt;
            bt.x = p0 ? base[off + t * 16]      : 0.0f;
            bt.y = p1 ? base[off + DP + t * 16] : 0.0f;
            acc[t] = __builtin_amdgcn_wmma_f32_16x16x4_f32(
                false, ones, false, bt, (short)0, acc[t], false, false);
        }
    }

    // D row M=0 (VGPR0 of C/D, lanes 0-15) holds the column sums; apply mean +
    // the (p,d) -> (d,p) output permutation on the tiny 4 MB store.
    const float cf = (float)cnt;
    if (lane < 16) {
        float* __restrict__ og = out + (size_t)g * DP;
        #pragma unroll
        for (int t = 0; t < TILES_PER_WAVE; ++t) {
            const int col = wbase + t * 16 + nl;             // memory-order index p*256+d
            const int dp  = ((col & 255) << 2) | (col >> 8); // output index d*4+p
            og[dp] = acc[t][0] / cf;
        }
    }
}

extern "C" void kernel_launch(void* const* d_in, const int* in_sizes, int n_in,
                              void* d_out, int out_size, void* d_ws, size_t ws_size,
                              hipStream_t stream) {
    (void)n_in; (void)out_size; (void)ws_size;
    const float* feat   = (const float*)d_in[0];
    const int*   counts = (const int*)d_in[1];
    const int    G      = in_sizes[1];          // 1024 graphs
    int* offs = (int*)d_ws;

    scan_counts_kernel<<<1, G_MAX, 0, stream>>>(counts, offs, G);
    seg_mean_wmma_kernel<<<G, 256, 0, stream>>>(feat, counts, offs, (float*)d_out);
}